// mowLSTM_31920196944179
// MI455X (gfx1250) — compile-verified
//
#include <hip/hip_runtime.h>

// ---------------------------------------------------------------------------
// Mixture-of-weights LSTM for MI455X (gfx1250, wave32, WMMA, async-to-LDS).
//   B=256, T=256, D=128, H=256, K=2, G=4H=1024, OUT=64
//
// Persistent kernel, 64 blocks x 256 threads (8 waves):
//   block (rb, jb): rb = 64-batch-row group (4), jb = 16-wide hc group (16).
//   Block owns gate tiles {bT = rb*4..rb*4+3} x {gT = jb + q*16, q=0..3},
//   i.e. all four gates (i,f,g,o) of its hc group -> cell update is
//   block-local, gates live in LDS, c lives in registers, ONE grid
//   barrier per timestep.
// Per step:
//   1. async-copy h slab (64x256) into LDS (global_load_async_to_lds_b128);
//      x_{t} was already prefetched into a double buffer last step
//   2. weight premix pass (b128-vectorized): lds_w = c0*W0 + c1*W1
//   3. s_wait_asynccnt 0; barrier; WMMA f32 16x16x4 over K=384 from LDS
//   4. barrier; LSTM cell via v_tanh_f32; h -> global, hs
//   5. prefetch x_{t+1} (async); grid barrier
// Final: data-parallel WMMA GEMM out = hs @ W_out + b_out -> (B,T,OUT)
// ---------------------------------------------------------------------------

typedef __attribute__((ext_vector_type(2))) float v2f;
typedef __attribute__((ext_vector_type(4))) float v4f;
typedef __attribute__((ext_vector_type(8))) float v8f;

#define T_ 256
#define B_ 256
#define D_ 128
#define H_ 256
#define G_ 1024
#define O_ 64
#define NBLK 64

// padded LDS strides (floats): bank-conflict-free fragment reads, 16B rows
#define XS 132   // x slab stride   (528 B, %16==0, bank step 4)
#define HS 260   // h slab stride   (1040 B, %16==0, bank step 4)
#define WS 388   // w slab stride   (1552 B, %16==0, bank step 4)
#define GS 68    // gate slab stride

// gfx1250 hardware tanh (TRANS op); trailing v_nop honors the TRANS
// co-execution hazard rule since the scheduler can't see through asm.
__device__ __forceinline__ float fast_tanh(float x) {
  float y;
  asm volatile("v_tanh_f32 %0, %1\n\tv_nop" : "=v"(y) : "v"(x));
  return y;
}
__device__ __forceinline__ float fast_sig(float x) {
  return 0.5f + 0.5f * fast_tanh(0.5f * x);
}

__device__ __forceinline__ v8f wmma_f32(v2f a, v2f b, v8f c) {
  return __builtin_amdgcn_wmma_f32_16x16x4_f32(false, a, false, b, (short)0, c,
                                               false, false);
}

// async 16-byte global -> LDS copy (tracked by ASYNCcnt)
__device__ __forceinline__ void async_ld_b128(unsigned lds_byte_off,
                                              const void* gaddr) {
  asm volatile("global_load_async_to_lds_b128 %0, %1, off"
               :: "v"(lds_byte_off), "v"(gaddr) : "memory");
}
__device__ __forceinline__ void wait_async0() {
  asm volatile("s_wait_asynccnt 0" ::: "memory");
}

__device__ __forceinline__ void grid_barrier(int* bar, int nblk) {
  __syncthreads();
  if (threadIdx.x == 0) {
    int gen = __hip_atomic_load(&bar[1], __ATOMIC_RELAXED, __HIP_MEMORY_SCOPE_AGENT);
    int prev = __hip_atomic_fetch_add(&bar[0], 1, __ATOMIC_ACQ_REL, __HIP_MEMORY_SCOPE_AGENT);
    if (prev == nblk - 1) {
      __hip_atomic_store(&bar[0], 0, __ATOMIC_RELAXED, __HIP_MEMORY_SCOPE_AGENT);
      __hip_atomic_fetch_add(&bar[1], 1, __ATOMIC_ACQ_REL, __HIP_MEMORY_SCOPE_AGENT);
    } else {
      while (__hip_atomic_load(&bar[1], __ATOMIC_ACQUIRE, __HIP_MEMORY_SCOPE_AGENT) == gen) {
        __builtin_amdgcn_s_sleep(2);
      }
    }
  }
  __syncthreads();
}

// coef[t,:] = softmax(alpha[t,:]); bt[t,g] = c0*b[0,g] + c1*b[1,g]
__global__ void coef_bt_kernel(const float* __restrict__ alpha,
                               const float* __restrict__ b,
                               float* __restrict__ coef, float* __restrict__ bt) {
  const int t = blockIdx.x;
  const float a0 = alpha[2 * t + 0];
  const float a1 = alpha[2 * t + 1];
  const float m = fmaxf(a0, a1);
  const float e0 = __expf(a0 - m);
  const float e1 = __expf(a1 - m);
  const float inv = 1.0f / (e0 + e1);
  const float c0 = e0 * inv, c1 = e1 * inv;
  if (threadIdx.x == 0) {
    coef[2 * t + 0] = c0;
    coef[2 * t + 1] = c1;
  }
  for (int g = threadIdx.x; g < G_; g += blockDim.x)
    bt[(size_t)t * G_ + g] = c0 * b[g] + c1 * b[G_ + g];
}

// out[k][c][r] = in[k][r][c]
__global__ void transpose_kernel(const float* __restrict__ in, float* __restrict__ out,
                                 int K, int R, int C) {
  const int n = K * R * C;
  for (int i = blockIdx.x * blockDim.x + threadIdx.x; i < n;
       i += gridDim.x * blockDim.x) {
    const int k = i / (R * C);
    const int rem = i - k * (R * C);
    const int r = rem / C;
    const int cc = rem - r * C;
    out[(size_t)k * R * C + (size_t)cc * R + r] = in[i];
  }
}

__global__ void init_kernel(float* __restrict__ h, int* __restrict__ bar) {
  const int i = blockIdx.x * blockDim.x + threadIdx.x;
  if (i < B_ * H_) h[i] = 0.0f;
  if (i < 2) bar[i] = 0;
}

__global__ void __launch_bounds__(256, 1)
lstm_kernel(const float* __restrict__ x, const float* __restrict__ WihT,
            const float* __restrict__ WhhT, const float* __restrict__ coef,
            const float* __restrict__ bt, float* __restrict__ h,
            float* __restrict__ hs, int* __restrict__ bar) {
  __shared__ float lds_w[64 * WS];       // premixed weights [col64][k384]
  __shared__ float lds_x[2 * 64 * XS];   // x slab double buffer [row64][d128]
  __shared__ float lds_h[64 * HS];       // h slab   [row64][k256]
  __shared__ float lds_g[64 * GS];       // gates    [row64][col64]
  __shared__ float lds_bt[64];           // premixed bias for block cols

  const int tid = threadIdx.x;
  const int lane = tid & 31;
  const int wid = tid >> 5;
  const int lo = lane & 15;
  const int hi = lane >> 4;
  const int rb = blockIdx.x >> 4;  // 0..3  : rows rb*64 .. rb*64+63
  const int jb = blockIdx.x & 15;  // 0..15 : hc cols jb*16 .. jb*16+15

  // weight-mix pass mapping: 4 threads per column, 96 contiguous k's each
  const int mcl = tid >> 2;                       // column 0..63
  const int sub = tid & 3;                        // quarter-column
  const int mq = mcl >> 4, mlo = mcl & 15;
  const int mgcol = mq * 256 + jb * 16 + mlo;     // column in G
  const float* wih0 = WihT + (size_t)mgcol * D_;
  const float* wih1 = WihT + (size_t)(G_ + mgcol) * D_;
  const float* whh0 = WhhT + (size_t)mgcol * H_;
  const float* whh1 = WhhT + (size_t)(G_ + mgcol) * H_;

  // bias column for tid < 64
  const int bgcol = (tid >> 4) * 256 + jb * 16 + (tid & 15);

  // cell state in registers: thread owns cells idx = tid + i*256
  float creg[4] = {0.0f, 0.0f, 0.0f, 0.0f};

  const unsigned lds_x_base = (unsigned)(size_t)&lds_x[0];
  const unsigned lds_h_base = (unsigned)(size_t)&lds_h[0];
  const float* hslab = h + (size_t)rb * 64 * H_;  // contiguous 64KB

  // ---- prefetch x_0 into buffer 0
  {
    #pragma unroll
    for (int it = 0; it < 8; ++it) {
      const int id = tid + it * 256;
      const int r = id >> 5;
      const int cchunk = id & 31;
      const float* gsrc = x + (size_t)(rb * 64 + r) * T_ * D_ + cchunk * 4;
      async_ld_b128(lds_x_base + (unsigned)(r * (XS * 4) + cchunk * 16), gsrc);
    }
  }

  for (int t = 0; t < T_; ++t) {
    const int xb = t & 1;

    // ---- 1. async stage h (16 x b128) into LDS
    #pragma unroll
    for (int it = 0; it < 16; ++it) {
      const int id = tid + it * 256;             // 0..4095 16B chunks
      const int r = id >> 6;                     // 64 chunks per 1KB row
      const int cchunk = id & 63;
      async_ld_b128(lds_h_base + (unsigned)(r * (HS * 4) + cchunk * 16),
                    (const char*)hslab + id * 16);
    }

    // ---- 2. premix weights into LDS (b128-vectorized, overlaps asyncs)
    const float c0 = coef[2 * t + 0];
    const float c1 = coef[2 * t + 1];
    if (tid < 64) lds_bt[tid] = bt[(size_t)t * G_ + bgcol];
    #pragma unroll 4
    for (int it = 0; it < 24; ++it) {
      const int kk = sub * 96 + it * 4;          // 0..383, 4-aligned
      const float* p0 = (kk < D_) ? (wih0 + kk) : (whh0 + (kk - D_));
      const float* p1 = (kk < D_) ? (wih1 + kk) : (whh1 + (kk - D_));
      v4f w0 = *(const v4f*)p0;
      v4f w1 = *(const v4f*)p1;
      v4f wm;
      #pragma unroll
      for (int j = 0; j < 4; ++j) wm[j] = c0 * w0[j] + c1 * w1[j];
      *(v4f*)&lds_w[mcl * WS + kk] = wm;
    }

    wait_async0();
    __syncthreads();

    // ---- 3. gates = bt + [x_t | h] @ Wmix   (WMMA f32 16x16x4, K=384)
    #pragma unroll
    for (int tt = 0; tt < 2; ++tt) {
      const int tau = wid * 2 + tt;   // 0..15
      const int bTl = tau & 3;        // local row tile
      const int q = tau >> 2;         // gate 0..3
      const int m = bTl * 16 + lo;    // local A row
      const int cl = q * 16 + lo;     // local B/C col

      v8f acc;
      const float bv = lds_bt[cl];
      #pragma unroll
      for (int j = 0; j < 8; ++j) acc[j] = bv;

      const float* wcol = &lds_w[cl * WS];
      const float* xrow = &lds_x[xb * (64 * XS) + m * XS];
      const float* hrow = &lds_h[m * HS];
      #pragma unroll 4
      for (int k = 0; k < D_; k += 4) {
        const int kb = k + 2 * hi;
        v2f a = *(const v2f*)(xrow + kb);
        v2f bb = *(const v2f*)(wcol + kb);
        acc = wmma_f32(a, bb, acc);
      }
      #pragma unroll 4
      for (int k = 0; k < H_; k += 4) {
        const int kb = k + 2 * hi;
        v2f a = *(const v2f*)(hrow + kb);
        v2f bb = *(const v2f*)(wcol + D_ + kb);
        acc = wmma_f32(a, bb, acc);
      }
      #pragma unroll
      for (int j = 0; j < 8; ++j)
        lds_g[(bTl * 16 + j + 8 * hi) * GS + cl] = acc[j];
    }

    __syncthreads();

    // ---- 4. LSTM cell via v_tanh_f32 (c stays in registers)
    #pragma unroll
    for (int i = 0; i < 4; ++i) {
      const int idx = tid + i * 256;   // 0..1023
      const int lr = idx >> 4;         // local row 0..63
      const int lc = idx & 15;         // local hc  0..15
      const float ig = lds_g[lr * GS + 0 + lc];
      const float fg = lds_g[lr * GS + 16 + lc];
      const float gg = lds_g[lr * GS + 32 + lc];
      const float og = lds_g[lr * GS + 48 + lc];
      const float cn = fast_sig(fg) * creg[i] + fast_sig(ig) * fast_tanh(gg);
      const float hn = fast_sig(og) * fast_tanh(cn);
      creg[i] = cn;
      const int gb = rb * 64 + lr;
      const int hc = jb * 16 + lc;
      h[(size_t)gb * H_ + hc] = hn;
      hs[(size_t)t * (B_ * H_) + (size_t)gb * H_ + hc] = hn;
    }

    // ---- 5. prefetch x_{t+1} into the other buffer (overlaps barrier)
    if (t + 1 < T_) {
      #pragma unroll
      for (int it = 0; it < 8; ++it) {
        const int id = tid + it * 256;
        const int r = id >> 5;
        const int cchunk = id & 31;
        const float* gsrc = x + (size_t)(rb * 64 + r) * T_ * D_ +
                            (size_t)(t + 1) * D_ + cchunk * 4;
        async_ld_b128(lds_x_base +
                          (unsigned)((xb ^ 1) * (64 * XS * 4) + r * (XS * 4) +
                                     cchunk * 16),
                      gsrc);
      }
    }

    // ---- 6. single grid-wide sync per timestep
    grid_barrier(bar, NBLK);
  }
}

// out[b,t,:] = hs[t,b,:] @ W_out + b_out ; hs as (T*B, H); 1 wave per tile
__global__ void out_kernel(const float* __restrict__ hs,
                           const float* __restrict__ WoutT,
                           const float* __restrict__ bout,
                           float* __restrict__ out) {
  const int lane = threadIdx.x & 31;
  const int wid = threadIdx.x >> 5;
  const int gwave = blockIdx.x * 8 + wid;  // 0..16383
  const int mT = gwave >> 2;
  const int nT = gwave & 3;
  const int lo = lane & 15;
  const int hi = lane >> 4;
  const int row = mT * 16 + lo;  // r = t*B + b
  const int col = nT * 16 + lo;

  const float* a0 = hs + (size_t)row * H_;
  const float* w = WoutT + (size_t)col * H_;
  v8f acc;
  const float bv = bout[col];
  #pragma unroll
  for (int j = 0; j < 8; ++j) acc[j] = bv;
  #pragma unroll 4
  for (int k = 0; k < H_; k += 4) {
    const int kb = k + 2 * hi;
    v2f a = *(const v2f*)(a0 + kb);
    v2f bb = *(const v2f*)(w + kb);
    acc = wmma_f32(a, bb, acc);
  }
  #pragma unroll
  for (int j = 0; j < 8; ++j) {
    const int r = mT * 16 + j + 8 * hi;
    const int tt = r >> 8;
    const int b = r & (B_ - 1);
    out[((size_t)b * T_ + tt) * O_ + col] = acc[j];
  }
}

extern "C" void kernel_launch(void* const* d_in, const int* in_sizes, int n_in,
                              void* d_out, int out_size, void* d_ws, size_t ws_size,
                              hipStream_t stream) {
  const float* x     = (const float*)d_in[0];  // (B,T,D)
  const float* W_ih  = (const float*)d_in[1];  // (K,D,G)
  const float* W_hh  = (const float*)d_in[2];  // (K,H,G)
  const float* b     = (const float*)d_in[3];  // (K,G)
  const float* alpha = (const float*)d_in[4];  // (T,K)
  const float* W_out = (const float*)d_in[5];  // (H,OUT)
  const float* b_out = (const float*)d_in[6];  // (OUT)
  float* out = (float*)d_out;                  // (B,T,OUT)

  float* ws = (float*)d_ws;
  size_t off = 0;
  float* coef  = ws + off; off += (size_t)T_ * 2;
  float* bt    = ws + off; off += (size_t)T_ * G_;
  float* WihT  = ws + off; off += (size_t)2 * G_ * D_;   // (K,G,D)
  float* WhhT  = ws + off; off += (size_t)2 * G_ * H_;   // (K,G,H)
  float* WoutT = ws + off; off += (size_t)O_ * H_;       // (OUT,H)
  float* hbuf  = ws + off; off += (size_t)B_ * H_;
  float* hs    = ws + off; off += (size_t)T_ * B_ * H_;
  int* bar = (int*)(ws + off);

  coef_bt_kernel<<<T_, 256, 0, stream>>>(alpha, b, coef, bt);
  transpose_kernel<<<512, 256, 0, stream>>>(W_ih, WihT, 2, D_, G_);
  transpose_kernel<<<1024, 256, 0, stream>>>(W_hh, WhhT, 2, H_, G_);
  transpose_kernel<<<64, 256, 0, stream>>>(W_out, WoutT, 1, H_, O_);
  init_kernel<<<(B_ * H_ + 255) / 256, 256, 0, stream>>>(hbuf, bar);
  lstm_kernel<<<NBLK, 256, 0, stream>>>(x, WihT, WhhT, coef, bt, hbuf, hs, bar);
  out_kernel<<<2048, 256, 0, stream>>>(hs, WoutT, b_out, out);
}